// CustomSIGLIP_55104430408114
// MI455X (gfx1250) — compile-verified
//
#include <hip/hip_runtime.h>
#include <math.h>

#define B_  16384
#define D_  1152
#define H_  288
#define ND_ 8
#define C_  751

typedef float v2f __attribute__((ext_vector_type(2)));
typedef float v8f __attribute__((ext_vector_type(8)));

__device__ __forceinline__ float gelu_exact(float x) {
    // torch nn.GELU default: exact erf form
    return 0.5f * x * (1.0f + erff(x * 0.70710678118654752440f));
}

// ---------------- bucketing ----------------
__global__ void siglip_bucket_init(int* __restrict__ counts) {
    if (threadIdx.x < ND_) counts[threadIdx.x] = 0;
}

__global__ void siglip_bucket_scatter(const int* __restrict__ dom,
                                      int* __restrict__ counts,
                                      int* __restrict__ buckets) {
    int b = blockIdx.x * blockDim.x + threadIdx.x;
    if (b < B_) {
        int d = dom[b];
        int pos = atomicAdd(&counts[d], 1);
        buckets[d * B_ + pos] = b;
    }
}

// ---------------- shared WMMA inner loop: 16(M) x 64(N) tile ----------------
// A fragment (16x4 f32): lane L holds row M = L&15; vgpr v holds K = 2*(L>=16)+v.
// B fragment (4x16 f32): lane L holds col N = L&15; vgpr v holds K = 2*(L>=16)+v.
// Branch-free: out-of-range columns are CLAMPED (their acc values are garbage
// duplicates of column Nlim-1 but the store epilogue masks them out).
__device__ __forceinline__ void wmma_tile_16x64(
    const float* __restrict__ Xrow,          // &X[arow * ldx] (per-lane row base)
    const float* __restrict__ W, int ldw,    // K x N weight matrix, row stride ldw
    int K, int Nlim, int n0, int lane, v8f acc[4])
{
    const int koff = (lane < 16) ? 0 : 2;
    const int ncol = lane & 15;
    const int nc0 = min(n0 + ncol,      Nlim - 1);
    const int nc1 = min(n0 + ncol + 16, Nlim - 1);
    const int nc2 = min(n0 + ncol + 32, Nlim - 1);
    const int nc3 = min(n0 + ncol + 48, Nlim - 1);

    const float* xp = Xrow + koff;
    const float* wp = W + (size_t)koff * ldw;

    v8f a0 = acc[0], a1 = acc[1], a2 = acc[2], a3 = acc[3];

#pragma unroll 2
    for (int k = 0; k < K; k += 4) {
        v2f av = *(const v2f*)(xp + k);
        const float* wk = wp + (size_t)k * ldw;
        __builtin_prefetch(wk + (size_t)8 * ldw + nc0, 0, 0);
        v2f b0, b1, b2, b3;
        b0.x = wk[nc0]; b0.y = wk[nc0 + ldw];
        b1.x = wk[nc1]; b1.y = wk[nc1 + ldw];
        b2.x = wk[nc2]; b2.y = wk[nc2 + ldw];
        b3.x = wk[nc3]; b3.y = wk[nc3 + ldw];
        a0 = __builtin_amdgcn_wmma_f32_16x16x4_f32(false, av, false, b0, (short)0, a0, false, false);
        a1 = __builtin_amdgcn_wmma_f32_16x16x4_f32(false, av, false, b1, (short)0, a1, false, false);
        a2 = __builtin_amdgcn_wmma_f32_16x16x4_f32(false, av, false, b2, (short)0, a2, false, false);
        a3 = __builtin_amdgcn_wmma_f32_16x16x4_f32(false, av, false, b3, (short)0, a3, false, false);
    }
    acc[0] = a0; acc[1] = a1; acc[2] = a2; acc[3] = a3;
}

// ---------------- GEMM 1: a = gelu(x @ W1[dom]), [cnt x D] @ [D x H] ----------------
__global__ void siglip_gemm1(const float* __restrict__ X,
                             const float* __restrict__ W1,
                             const int* __restrict__ counts,
                             const int* __restrict__ buckets,
                             float* __restrict__ aout) {
    const int dom  = blockIdx.y;
    const int cnt  = counts[dom];
    const int tile = blockIdx.x;
    if (tile * 16 >= cnt) return;
    const int lane = threadIdx.x;
    const int* bk  = buckets + dom * B_;
    const int ncol = lane & 15;
    const int arow = bk[min(tile * 16 + ncol, cnt - 1)];
    const int n0   = blockIdx.z * 64;

    v8f acc[4] = {};
    wmma_tile_16x64(X + (size_t)arow * D_, W1 + (size_t)dom * D_ * H_, H_,
                    D_, H_, n0, lane, acc);

    const int mbase = tile * 16 + ((lane < 16) ? 0 : 8);
#pragma unroll
    for (int t = 0; t < 4; ++t) {
        int n = n0 + ncol + 16 * t;
        if (n >= H_) continue;
#pragma unroll
        for (int v = 0; v < 8; ++v) {
            int p = mbase + v;
            if (p < cnt) {
                int r = bk[p];
                aout[(size_t)r * H_ + n] = gelu_exact(acc[t][v]);
            }
        }
    }
}

// ---- GEMM 2: mixed = 0.4*gelu(a @ W2[dom]) + 0.6*x, [cnt x H] @ [H x D] ----
__global__ void siglip_gemm2(const float* __restrict__ A,
                             const float* __restrict__ W2,
                             const float* __restrict__ Xbase,
                             const int* __restrict__ counts,
                             const int* __restrict__ buckets,
                             float* __restrict__ mixed) {
    const int dom  = blockIdx.y;
    const int cnt  = counts[dom];
    const int tile = blockIdx.x;
    if (tile * 16 >= cnt) return;
    const int lane = threadIdx.x;
    const int* bk  = buckets + dom * B_;
    const int ncol = lane & 15;
    const int arow = bk[min(tile * 16 + ncol, cnt - 1)];
    const int n0   = blockIdx.z * 64;

    v8f acc[4] = {};
    wmma_tile_16x64(A + (size_t)arow * H_, W2 + (size_t)dom * H_ * D_, D_,
                    H_, D_, n0, lane, acc);

    const int mbase = tile * 16 + ((lane < 16) ? 0 : 8);
#pragma unroll
    for (int t = 0; t < 4; ++t) {
        int n = n0 + ncol + 16 * t;
        if (n >= D_) continue;
#pragma unroll
        for (int v = 0; v < 8; ++v) {
            int p = mbase + v;
            if (p < cnt) {
                int r = bk[p];
                float ad = gelu_exact(acc[t][v]);
                mixed[(size_t)r * D_ + n] =
                    0.4f * ad + 0.6f * Xbase[(size_t)r * D_ + n];
            }
        }
    }
}

// ---- GEMM 3: cls = mixed @ Wc[dom] + bc[dom], [cnt x D] @ [D x C] ----
__global__ void siglip_gemm3(const float* __restrict__ M,
                             const float* __restrict__ Wc,
                             const float* __restrict__ bc,
                             const int* __restrict__ counts,
                             const int* __restrict__ buckets,
                             float* __restrict__ cls) {
    const int dom  = blockIdx.y;
    const int cnt  = counts[dom];
    const int tile = blockIdx.x;
    if (tile * 16 >= cnt) return;
    const int lane = threadIdx.x;
    const int* bk  = buckets + dom * B_;
    const int ncol = lane & 15;
    const int arow = bk[min(tile * 16 + ncol, cnt - 1)];
    const int n0   = blockIdx.z * 64;

    v8f acc[4] = {};
    wmma_tile_16x64(M + (size_t)arow * D_, Wc + (size_t)dom * D_ * C_, C_,
                    D_, C_, n0, lane, acc);

    const int mbase = tile * 16 + ((lane < 16) ? 0 : 8);
#pragma unroll
    for (int t = 0; t < 4; ++t) {
        int n = n0 + ncol + 16 * t;
        if (n >= C_) continue;
        float bias = bc[dom * C_ + n];
#pragma unroll
        for (int v = 0; v < 8; ++v) {
            int p = mbase + v;
            if (p < cnt) {
                int r = bk[p];
                cls[(size_t)r * C_ + n] = acc[t][v] + bias;
            }
        }
    }
}

// ---------------- row-wise L2 normalize ----------------
__global__ void siglip_normalize(const float* __restrict__ mixed,
                                 float* __restrict__ out) {
    const int row = blockIdx.x;
    const float* x = mixed + (size_t)row * D_;
    float s = 0.0f;
    for (int i = threadIdx.x; i < D_; i += blockDim.x) {
        float v = x[i];
        s += v * v;
    }
    __shared__ float red[8];
#pragma unroll
    for (int off = 16; off > 0; off >>= 1) s += __shfl_down(s, off, 32);
    if ((threadIdx.x & 31) == 0) red[threadIdx.x >> 5] = s;
    __syncthreads();
    if (threadIdx.x == 0) {
        float t = 0.0f;
#pragma unroll
        for (int w = 0; w < 8; ++w) t += red[w];
        red[0] = t;
    }
    __syncthreads();
    const float inv = 1.0f / fmaxf(sqrtf(red[0]), 1e-6f);
    for (int i = threadIdx.x; i < D_; i += blockDim.x)
        out[(size_t)row * D_ + i] = x[i] * inv;
}

extern "C" void kernel_launch(void* const* d_in, const int* in_sizes, int n_in,
                              void* d_out, int out_size, void* d_ws, size_t ws_size,
                              hipStream_t stream) {
    const float* base    = (const float*)d_in[0];   // [B, D]
    const int*   domains = (const int*)  d_in[1];   // [B]
    const float* W1      = (const float*)d_in[2];   // [ND, D, H]
    const float* W2      = (const float*)d_in[3];   // [ND, H, D]
    const float* Wc      = (const float*)d_in[4];   // [ND, D, C]
    const float* bc      = (const float*)d_in[5];   // [ND, C]

    float* cls   = (float*)d_out;                   // [B, C]
    float* mnorm = cls + (size_t)B_ * C_;           // [B, D]

    char* ws = (char*)d_ws;
    int*   counts  = (int*)ws;                                   // 8 ints
    int*   buckets = (int*)(ws + 256);                           // ND*B ints
    float* a       = (float*)(ws + 256 + sizeof(int) * ND_ * B_);// [B, H]
    float* mixed   = a + (size_t)B_ * H_;                        // [B, D]

    siglip_bucket_init<<<1, 32, 0, stream>>>(counts);
    siglip_bucket_scatter<<<(B_ + 255) / 256, 256, 0, stream>>>(domains, counts, buckets);

    const int maxTiles = (B_ + 15) / 16; // worst case: one domain owns all samples
    dim3 g1(maxTiles, ND_, (H_ + 63) / 64);
    siglip_gemm1<<<g1, 32, 0, stream>>>(base, W1, counts, buckets, a);

    dim3 g2(maxTiles, ND_, (D_ + 63) / 64);
    siglip_gemm2<<<g2, 32, 0, stream>>>(a, W2, base, counts, buckets, mixed);

    dim3 g3(maxTiles, ND_, (C_ + 63) / 64);
    siglip_gemm3<<<g3, 32, 0, stream>>>(mixed, Wc, bc, counts, buckets, cls);

    siglip_normalize<<<B_, 256, 0, stream>>>(mixed, mnorm);
}